// DeepFM_16406775070798
// MI455X (gfx1250) — compile-verified
//
#include <hip/hip_runtime.h>
#include <hip/hip_bf16.h>

// ---------------- problem constants ----------------
#define N_TOT      131072
#define NDENSE     13
#define NSPARSE    26
#define NFIELD     39
#define EMB        16
#define VOCAB      100000
#define HID        32
#define DDIM       (NFIELD*EMB)        // 624
#define EPSV       1e-5f

#define TILE_M     16                  // samples per block
#define NBLK       (N_TOT/TILE_M)      // 8192
#define BLK1       256                 // pass1 block (8 waves)
#define DSTR       625                 // padded LDS row stride (floats)

typedef float v2f __attribute__((ext_vector_type(2)));
typedef float v8f __attribute__((ext_vector_type(8)));

// ---------------- workspace layout (float offsets) ----------------
#define WS_Z1      0L
#define WS_Z2      (WS_Z1 + (long)N_TOT*HID)
#define WS_P       (WS_Z2 + (long)N_TOT*HID)
#define WS_PART1   (WS_P  + (long)N_TOT)
#define WS_PART2   (WS_PART1 + (long)NBLK*64)
#define WS_A1      (WS_PART2 + (long)NBLK*64)
#define WS_C1      (WS_A1 + HID)
#define WS_A2      (WS_C1 + HID)
#define WS_CONST2  (WS_A2 + HID)

// =================================================================
// Pass 1: embeddings + FM term + z1 = deep @ Wl1 + bl1 (f32 WMMA)
// =================================================================
__global__ __launch_bounds__(BLK1)
void pass1_kernel(const float* __restrict__ Xi_dense,
                  const int*   __restrict__ Xi_sparse,
                  const float* __restrict__ Xv,
                  const float* __restrict__ bias,
                  const float* __restrict__ W1d, const float* __restrict__ b1d,
                  const float* __restrict__ T1,
                  const float* __restrict__ W2d, const float* __restrict__ b2d,
                  const float* __restrict__ T2,
                  const float* __restrict__ Wl1, const float* __restrict__ bl1,
                  float* __restrict__ ws)
{
    __shared__ float deepS[TILE_M * DSTR];     // second-embedding tile (16 x 624)
    __shared__ float sArr [TILE_M * EMB];      // per-sample field-sum s_k
    __shared__ float qArr [TILE_M * EMB];      // per-(sample,k) sum of sec^2
    __shared__ float frArr[TILE_M * EMB];      // per-(sample,k) sum of first

    const int tid = threadIdx.x;
    const int n0  = blockIdx.x * TILE_M;

    // ---- phase 1: one thread per (sample, emb-dim); loop over 39 fields ----
    {
        const int nl = tid >> 4;        // 0..15  (BLK1 == 256 == 16*16)
        const int k  = tid & 15;        // 0..15
        const int n  = n0 + nl;
        float s = 0.f, q = 0.f, fr = 0.f;
        #pragma unroll 1
        for (int f = 0; f < NFIELD; ++f) {
            const float v = Xv[n*NFIELD + f];
            float sec, fir;
            if (f < NDENSE) {
                const float x = Xi_dense[n*NDENSE + f];
                sec = (x * W2d[f*EMB + k] + b2d[f*EMB + k]) * v;
                fir = (x * W1d[f*EMB + k] + b1d[f*EMB + k]) * v;
            } else {
                const int  sf  = f - NDENSE;
                const long idx = (long)Xi_sparse[n*NSPARSE + sf];
                sec = T2[((long)sf*VOCAB + idx)*EMB + k] * v;   // 16 lanes -> 64B coalesced
                fir = T1[((long)sf*VOCAB + idx)*EMB + k] * v;
            }
            deepS[nl*DSTR + f*EMB + k] = sec;
            s += sec; q += sec*sec; fr += fir;
        }
        sArr [nl*EMB + k] = s;
        qArr [nl*EMB + k] = q;
        frArr[nl*EMB + k] = fr;
    }
    __syncthreads();

    // ---- wave 1: per-sample scalar p = first.sum + fm2.sum + bias ----
    if (tid >= 32 && tid < 32 + TILE_M) {
        const int nl = tid - 32;
        float ss = 0.f, qq = 0.f, ff = 0.f;
        #pragma unroll
        for (int k = 0; k < EMB; ++k) {
            const float s = sArr[nl*EMB + k];
            ss += s*s;
            qq += qArr [nl*EMB + k];
            ff += frArr[nl*EMB + k];
        }
        ws[WS_P + n0 + nl] = ff + 0.5f*(ss - qq) + bias[n0 + nl];
    }

    // ---- wave 0: z1 tile (16x32) = deepS (16x624) @ Wl1 (624x32) + bl1 ----
    if (tid < 32) {
        const int lane = tid;
        const int col  = lane & 15;     // A: M row / B,D: N column (mod 16)
        const int half = lane >> 4;     // K/M split across lane halves
        v8f acc0, acc1;
        #pragma unroll
        for (int v = 0; v < 8; ++v) { acc0[v] = bl1[col]; acc1[v] = bl1[col + 16]; }

        #pragma unroll 1
        for (int k = 0; k < DDIM; k += 4) {
            const int kk = k + 2*half;
            v2f a;  a.x  = deepS[col*DSTR + kk];        a.y  = deepS[col*DSTR + kk + 1];
            v2f b0; b0.x = Wl1[kk*HID + col];           b0.y = Wl1[(kk+1)*HID + col];
            v2f b1; b1.x = Wl1[kk*HID + col + 16];      b1.y = Wl1[(kk+1)*HID + col + 16];
            acc0 = __builtin_amdgcn_wmma_f32_16x16x4_f32(false, a, false, b0, (short)0, acc0, false, false);
            acc1 = __builtin_amdgcn_wmma_f32_16x16x4_f32(false, a, false, b1, (short)0, acc1, false, false);
        }

        // store z1 + deterministic per-block column partials (sum, sum^2)
        float s0 = 0.f, q0 = 0.f, s1 = 0.f, q1 = 0.f;
        #pragma unroll
        for (int v = 0; v < 8; ++v) {
            const int m = v + 8*half;
            const float d0 = acc0[v], d1 = acc1[v];
            ws[WS_Z1 + (long)(n0 + m)*HID + col]      = d0;
            ws[WS_Z1 + (long)(n0 + m)*HID + col + 16] = d1;
            s0 += d0; q0 += d0*d0; s1 += d1; q1 += d1*d1;
        }
        s0 += __shfl_xor(s0, 16, 32);
        q0 += __shfl_xor(q0, 16, 32);
        s1 += __shfl_xor(s1, 16, 32);
        q1 += __shfl_xor(q1, 16, 32);
        if (half == 0) {
            float* part = ws + WS_PART1 + (long)blockIdx.x*64;
            part[col]          = s0;
            part[col + 16]     = s1;
            part[32 + col]     = q0;
            part[32 + col +16] = q1;
        }
    }
}

// =================================================================
// Stats 1: reduce partials -> fold BN1 into affine h1 = a1*z1 + c1
// =================================================================
__global__ __launch_bounds__(32)
void stats1_kernel(const float* __restrict__ g1, const float* __restrict__ be1,
                   float* __restrict__ ws)
{
    const int c = threadIdx.x;          // 32 threads, one per hidden unit
    float s = 0.f, q = 0.f;
    const float* part = ws + WS_PART1;
    for (int b = 0; b < NBLK; ++b) { s += part[(long)b*64 + c]; q += part[(long)b*64 + 32 + c]; }
    const float m   = s * (1.0f / (float)N_TOT);
    const float var = q * (1.0f / (float)N_TOT) - m*m;
    const float a   = g1[c] * rsqrtf(var + EPSV);
    ws[WS_A1 + c] = a;
    ws[WS_C1 + c] = be1[c] - m*a;
}

// =================================================================
// Pass 2: z2 = (a1*z1 + c1) @ Wl2 + bl2   (f32 WMMA, K = 32)
// =================================================================
__global__ __launch_bounds__(32)
void pass2_kernel(const float* __restrict__ Wl2, const float* __restrict__ bl2,
                  float* __restrict__ ws)
{
    const int lane = threadIdx.x;
    const int col  = lane & 15;
    const int half = lane >> 4;
    const long n0  = (long)blockIdx.x * TILE_M;
    const float* a1 = ws + WS_A1;
    const float* c1 = ws + WS_C1;
    const float* z1 = ws + WS_Z1;

    v8f acc0, acc1;
    #pragma unroll
    for (int v = 0; v < 8; ++v) { acc0[v] = bl2[col]; acc1[v] = bl2[col + 16]; }

    #pragma unroll
    for (int k = 0; k < HID; k += 4) {
        const int kk = k + 2*half;
        v2f a;
        a.x = a1[kk]   * z1[(n0 + col)*HID + kk]     + c1[kk];
        a.y = a1[kk+1] * z1[(n0 + col)*HID + kk + 1] + c1[kk+1];
        v2f b0; b0.x = Wl2[kk*HID + col];      b0.y = Wl2[(kk+1)*HID + col];
        v2f b1; b1.x = Wl2[kk*HID + col + 16]; b1.y = Wl2[(kk+1)*HID + col + 16];
        acc0 = __builtin_amdgcn_wmma_f32_16x16x4_f32(false, a, false, b0, (short)0, acc0, false, false);
        acc1 = __builtin_amdgcn_wmma_f32_16x16x4_f32(false, a, false, b1, (short)0, acc1, false, false);
    }

    float s0 = 0.f, q0 = 0.f, s1 = 0.f, q1 = 0.f;
    #pragma unroll
    for (int v = 0; v < 8; ++v) {
        const int m = v + 8*half;
        const float d0 = acc0[v], d1 = acc1[v];
        ws[WS_Z2 + (n0 + m)*HID + col]      = d0;
        ws[WS_Z2 + (n0 + m)*HID + col + 16] = d1;
        s0 += d0; q0 += d0*d0; s1 += d1; q1 += d1*d1;
    }
    s0 += __shfl_xor(s0, 16, 32);
    q0 += __shfl_xor(q0, 16, 32);
    s1 += __shfl_xor(s1, 16, 32);
    q1 += __shfl_xor(q1, 16, 32);
    if (half == 0) {
        float* part = ws + WS_PART2 + (long)blockIdx.x*64;
        part[col]           = s0;
        part[col + 16]      = s1;
        part[32 + col]      = q0;
        part[32 + col + 16] = q1;
    }
}

// =================================================================
// Stats 2: BN2 folded -> h2.sum = sum_c a2[c]*z2[c] + C2
// =================================================================
__global__ __launch_bounds__(32)
void stats2_kernel(const float* __restrict__ g2, const float* __restrict__ be2,
                   float* __restrict__ ws)
{
    __shared__ float red[32];
    const int c = threadIdx.x;
    float s = 0.f, q = 0.f;
    const float* part = ws + WS_PART2;
    for (int b = 0; b < NBLK; ++b) { s += part[(long)b*64 + c]; q += part[(long)b*64 + 32 + c]; }
    const float m   = s * (1.0f / (float)N_TOT);
    const float var = q * (1.0f / (float)N_TOT) - m*m;
    const float a   = g2[c] * rsqrtf(var + EPSV);
    ws[WS_A2 + c] = a;
    red[c] = be2[c] - m*a;
    __syncthreads();
    if (c == 0) {
        float t = 0.f;
        for (int i = 0; i < 32; ++i) t += red[i];
        ws[WS_CONST2] = t;
    }
}

// =================================================================
// Final: out[n] = p[n] + sum_c a2[c]*z2[n,c] + C2
// =================================================================
__global__ __launch_bounds__(256)
void final_kernel(float* __restrict__ out, const float* __restrict__ ws)
{
    const long n = (long)blockIdx.x*256 + threadIdx.x;
    const float* a2 = ws + WS_A2;
    const float* z2 = ws + WS_Z2 + n*HID;
    float t = ws[WS_P + n] + ws[WS_CONST2];
    #pragma unroll
    for (int c = 0; c < HID; ++c) t += a2[c]*z2[c];
    out[n] = t;
}

// =================================================================
extern "C" void kernel_launch(void* const* d_in, const int* in_sizes, int n_in,
                              void* d_out, int out_size, void* d_ws, size_t ws_size,
                              hipStream_t stream)
{
    const float* Xi_dense  = (const float*)d_in[0];
    const int*   Xi_sparse = (const int*)  d_in[1];
    const float* Xv        = (const float*)d_in[2];
    const float* bias      = (const float*)d_in[3];
    const float* W1d       = (const float*)d_in[4];
    const float* b1d       = (const float*)d_in[5];
    const float* T1        = (const float*)d_in[6];
    const float* W2d       = (const float*)d_in[7];
    const float* b2d       = (const float*)d_in[8];
    const float* T2        = (const float*)d_in[9];
    const float* Wl1       = (const float*)d_in[10];
    const float* bl1       = (const float*)d_in[11];
    const float* g1        = (const float*)d_in[12];
    const float* be1       = (const float*)d_in[13];
    const float* Wl2       = (const float*)d_in[14];
    const float* bl2       = (const float*)d_in[15];
    const float* g2        = (const float*)d_in[16];
    const float* be2       = (const float*)d_in[17];
    float* ws  = (float*)d_ws;
    float* out = (float*)d_out;

    pass1_kernel<<<NBLK, BLK1, 0, stream>>>(Xi_dense, Xi_sparse, Xv, bias,
                                            W1d, b1d, T1, W2d, b2d, T2,
                                            Wl1, bl1, ws);
    stats1_kernel<<<1, 32, 0, stream>>>(g1, be1, ws);
    pass2_kernel<<<NBLK, 32, 0, stream>>>(Wl2, bl2, ws);
    stats2_kernel<<<1, 32, 0, stream>>>(g2, be2, ws);
    final_kernel<<<N_TOT/256, 256, 0, stream>>>(out, ws);
}